// Agent_57732950393167
// MI455X (gfx1250) — compile-verified
//
#include <hip/hip_runtime.h>
#include <hip/hip_bf16.h>
#include <math.h>

// ---------------------------------------------------------------------------
// Problem constants (from reference): B=32, N=1024, D=128, H=8, V=8
// T = B*V = 256 total "vehicles". KS = D/H = 16. F_V=4, F_ND=8.
// ---------------------------------------------------------------------------
#define Bn   32
#define Nn   1024
#define Dn   128
#define Hn   8
#define Vn   8
#define KSn  16
#define Tn   (Bn * Vn)      // 256
#define FND  8
#define FV   4
#define NEG_BIG (-1e30f)

typedef __attribute__((ext_vector_type(2))) float v2f;
typedef __attribute__((ext_vector_type(8))) float v8f;
typedef __attribute__((ext_vector_type(4))) unsigned int u32x4;
typedef __attribute__((ext_vector_type(8))) int i32x8;
typedef __attribute__((ext_vector_type(4))) int i32x4;

// ---------------------------------------------------------------------------
// TDM helper: issue TENSOR_LOAD_TO_LDS of a 2-D f32 tile
//   rows x rowlen elements, global row stride = row_stride elements.
// Descriptor layout per CDNA5 ISA §8.3/§8.4 (D# groups 0 and 1).
// All inputs must be wave-uniform (SGPR-resident).
// ---------------------------------------------------------------------------
__device__ __forceinline__ void tdm_load_tile_f32(unsigned lds_off,
                                                  const float* gsrc,
                                                  unsigned rows, unsigned rowlen,
                                                  unsigned row_stride) {
    unsigned long long ga = (unsigned long long)(uintptr_t)gsrc;
    u32x4 g0;
    g0[0] = 1u;                                        // count=1 (valid D#)
    g0[1] = lds_off;                                   // lds_addr (bytes)
    g0[2] = (unsigned)(ga & 0xFFFFFFFFu);              // global_addr[31:0]
    g0[3] = (unsigned)((ga >> 32) & 0x01FFFFFFu)       // global_addr[56:32]
          | (2u << 30);                                // type=2 ("image")
    i32x8 g1;
    g1[0] = (int)(2u << 16);                           // data_size=2 -> 4 bytes
    g1[1] = (int)((rowlen & 0xFFFFu) << 16);           // tensor_dim0[15:0]
    g1[2] = (int)(((rowlen >> 16) & 0xFFFFu)           // tensor_dim0[31:16]
          | ((rows & 0xFFFFu) << 16));                 // tensor_dim1[15:0]
    g1[3] = (int)(((rows >> 16) & 0xFFFFu)             // tensor_dim1[31:16]
          | ((rowlen & 0xFFFFu) << 16));               // tile_dim0
    g1[4] = (int)(rows & 0xFFFFu);                     // tile_dim1 (tile_dim2=0)
    g1[5] = (int)row_stride;                           // tensor_dim0_stride[31:0]
    g1[6] = 0;                                         // stride0 hi / stride1 lo
    g1[7] = 0;                                         // tensor_dim1_stride hi
    i32x4 z4 = {0, 0, 0, 0};
#if defined(__clang_major__) && (__clang_major__ >= 23)
    i32x8 z8 = {0, 0, 0, 0, 0, 0, 0, 0};
    __builtin_amdgcn_tensor_load_to_lds(g0, g1, z4, z4, z8, 0);
#else
    __builtin_amdgcn_tensor_load_to_lds(g0, g1, z4, z4, 0);
#endif
}

// ---------------------------------------------------------------------------
// Kernel 1: per-t setup.
//   cur[t,:]   = concat(fc[b], vdf[t]) @ W_cs          (132 -> 128)
//   query[t,:] = fc[b] + cur[t]
//   wK[t,h,f]  = sum_k query[t, h*16+k] * W_ns[f, 128 + h*16 + k]
// (wK is the rank-1 refactor of the K-dynamic attention term)
// ---------------------------------------------------------------------------
__global__ void k_setup(const float* __restrict__ fc,
                        const float* __restrict__ vdf,
                        const float* __restrict__ W_cs,
                        const float* __restrict__ W_ns,
                        float* __restrict__ query,
                        float* __restrict__ wK) {
    const int t = blockIdx.x;
    const int b = t >> 3;          // t / V
    const int d = threadIdx.x;     // 0..127
    __shared__ float q_sh[Dn];

    const float* fcb = fc + b * Dn;
    float acc = 0.f;
    #pragma unroll 4
    for (int i = 0; i < Dn; ++i) acc += fcb[i] * W_cs[i * Dn + d];
    const float* vt = vdf + t * FV;
    #pragma unroll
    for (int j = 0; j < FV; ++j) acc += vt[j] * W_cs[(Dn + j) * Dn + d];

    const float q = fcb[d] + acc;           // query = fc_rep + cur
    q_sh[d] = q;
    query[t * Dn + d] = q;
    __syncthreads();

    if (d < Hn * FND) {                     // 64 (h,f) pairs
        const int h = d >> 3, f = d & 7;
        const float* wn = W_ns + f * (3 * Dn) + Dn + h * KSn; // K-block of W_ns
        const float* qh = q_sh + h * KSn;
        float s = 0.f;
        #pragma unroll
        for (int k = 0; k < KSn; ++k) s += qh[k] * wn[k];
        wK[t * 64 + h * 8 + f] = s;
    }
}

// ---------------------------------------------------------------------------
// Kernel 2: attention. One block per t (256 threads = 8 wave32), wave h does
// head h over all N=1024 nodes (32 nodes/lane).
//   compat = (Q.gK_static + ndf.wK) / 4 + log_mask
//   softmax over N (two-pass with compat cached in LDS, wave shfl reductions)
//   heads[t, h*16+k] = (sum_n e * gV)/S + ((sum_n e * ndf)/S) . W_V_head
// ---------------------------------------------------------------------------
__global__ void k_attn(const float* __restrict__ query,
                       const float* __restrict__ wK,
                       const float* __restrict__ ndf,
                       const float* __restrict__ gK,
                       const float* __restrict__ gV,
                       const unsigned char* __restrict__ maskp,
                       const float* __restrict__ W_ns,
                       float* __restrict__ heads) {
    const int t = blockIdx.x;
    const int b = t >> 3;
    const int h    = threadIdx.x >> 5;   // wave id == head
    const int lane = threadIdx.x & 31;

    __shared__ float compat[Hn][Nn];     // 32 KB

    float Q[KSn], wk[FND];
    #pragma unroll
    for (int k = 0; k < KSn; ++k) Q[k] = query[t * Dn + h * KSn + k];
    #pragma unroll
    for (int f = 0; f < FND; ++f) wk[f] = wK[t * 64 + h * 8 + f];

    const float* gKb = gK + ((size_t)(h * Bn + b)) * Nn * KSn;
    const float* gVb = gV + ((size_t)(h * Bn + b)) * Nn * KSn;
    const float* ndb = ndf + (size_t)t * Nn * FND;
    const unsigned char* mb = maskp + (size_t)t * Nn;

    // pass 1: compat + running max
    float mx = -3.4e38f;
    for (int n = lane; n < Nn; n += 32) {
        const float* kp = gKb + n * KSn;
        const float* np = ndb + n * FND;
        float c = 0.f;
        #pragma unroll
        for (int k = 0; k < KSn; ++k) c += Q[k] * kp[k];
        #pragma unroll
        for (int f = 0; f < FND; ++f) c += wk[f] * np[f];
        c *= 0.25f;                              // 1/sqrt(KS)
        if (!mb[n]) c = NEG_BIG;                 // + log(mask)
        compat[h][n] = c;
        mx = fmaxf(mx, c);
    }
    #pragma unroll
    for (int off = 16; off > 0; off >>= 1)
        mx = fmaxf(mx, __shfl_xor(mx, off, 32));

    // pass 2: exp-sum + weighted accumulation
    float sum = 0.f;
    float hacc[KSn], sacc[FND];
    #pragma unroll
    for (int k = 0; k < KSn; ++k) hacc[k] = 0.f;
    #pragma unroll
    for (int f = 0; f < FND; ++f) sacc[f] = 0.f;

    for (int n = lane; n < Nn; n += 32) {
        const float e = __expf(compat[h][n] - mx);
        sum += e;
        const float* vp = gVb + n * KSn;
        const float* np = ndb + n * FND;
        #pragma unroll
        for (int k = 0; k < KSn; ++k) hacc[k] += e * vp[k];
        #pragma unroll
        for (int f = 0; f < FND; ++f) sacc[f] += e * np[f];
    }
    #pragma unroll
    for (int off = 16; off > 0; off >>= 1) {
        sum += __shfl_xor(sum, off, 32);
        #pragma unroll
        for (int k = 0; k < KSn; ++k) hacc[k] += __shfl_xor(hacc[k], off, 32);
        #pragma unroll
        for (int f = 0; f < FND; ++f) sacc[f] += __shfl_xor(sacc[f], off, 32);
    }

    if (lane < KSn) {
        const float inv = 1.f / sum;
        float hv = hacc[lane] * inv;
        #pragma unroll
        for (int f = 0; f < FND; ++f)       // dynamic-V correction (W_ns V-block)
            hv += (sacc[f] * inv) * W_ns[f * (3 * Dn) + h * KSn + lane];
        heads[t * Dn + h * KSn + lane] = hv;
    }
}

// ---------------------------------------------------------------------------
// Kernel 3: final_Q[t,:] = heads[t,:] @ W_out ; u[t,f] = W_ns_L[f,:] . fQ[t,:]
// ---------------------------------------------------------------------------
__global__ void k_final(const float* __restrict__ heads,
                        const float* __restrict__ W_out,
                        const float* __restrict__ W_ns,
                        float* __restrict__ fQ,
                        float* __restrict__ u) {
    const int t = blockIdx.x;
    const int d = threadIdx.x;   // 0..127
    __shared__ float h_sh[Dn];
    __shared__ float fq_sh[Dn];
    h_sh[d] = heads[t * Dn + d];
    __syncthreads();

    float acc = 0.f;
    #pragma unroll 4
    for (int e = 0; e < Dn; ++e) acc += h_sh[e] * W_out[e * Dn + d];
    fq_sh[d] = acc;
    fQ[t * Dn + d] = acc;
    __syncthreads();

    if (d < FND) {
        const float* wl = W_ns + d * (3 * Dn) + 2 * Dn;   // L-block of W_ns
        float s = 0.f;
        #pragma unroll 4
        for (int e = 0; e < Dn; ++e) s += wl[e] * fq_sh[e];
        u[t * FND + d] = s;
    }
}

// ---------------------------------------------------------------------------
// Kernel 4: logits via WMMA fed by the Tensor Data Mover.
// Grid = B*4 blocks, 64 threads (2 wave32) each. Block (b, quarter) owns 16
// node-tiles; each wave owns 8 tiles with 2 LDS tile buffers (double buffer).
// Per tile: TDM loads logit_K_static[b, n0:n0+16, 0:128] (16x128 f32, row
// stride 128) into LDS (TENSORcnt / s_wait_tensorcnt), then 32 x
// V_WMMA_F32_16X16X4_F32 accumulate the 16x16 logits tile (K=128).
// A layout  (16x4 fp32): lanes 0-15 hold {K=kk,kk+1} of row M=lane,
//                        lanes 16-31 hold {K=kk+2,kk+3} of row M=lane-16.
// B layout  (4x16 fp32): lane gives column N; VGPR pair gives same K split.
// C/D layout: VGPR i, lanes 0-15 -> M=i ; lanes 16-31 -> M=i+8 ; N = lane&15.
// Epilogue adds rank-1 dynamic term ndf.u, scales 1/sqrt(D), tanh*10, mask.
// ---------------------------------------------------------------------------
__global__ void k_logits(const float* __restrict__ fQ,
                         const float* __restrict__ u,
                         const float* __restrict__ lKs,
                         const float* __restrict__ ndf,
                         const unsigned char* __restrict__ maskp,
                         float* __restrict__ logits) {
    const int b       = blockIdx.x >> 2;
    const int quarter = blockIdx.x & 3;
    const int tid     = threadIdx.x;          // 0..63 (2 waves)

    __shared__ float A[16][Dn];               // 8 KB, rows 8..15 zero padding
    __shared__ float uLds[Vn][FND];
    __shared__ float Btile[2][2][16 * Dn];    // [wave][buf][n_local*128 + d] 32 KB

    for (int i = tid; i < 16 * Dn; i += 64) {
        const int m = i >> 7, dd = i & 127;
        A[m][dd] = (m < Vn) ? fQ[(b * Vn + m) * Dn + dd] : 0.f;
    }
    if (tid < Vn * FND)
        uLds[tid >> 3][tid & 7] = u[(b * Vn + (tid >> 3)) * FND + (tid & 7)];
    __syncthreads();

    const int wv   = __builtin_amdgcn_readfirstlane(tid >> 5);  // wave id (SGPR)
    const int lane = tid & 31;
    const int half = lane >> 4;               // 0: lanes 0-15, 1: lanes 16-31
    const int l16  = lane & 15;
    const float invsD = 0.08838834764831845f; // 1/sqrt(128)

    const int jbase = quarter * 16 + wv * 8;  // this wave's first node-tile
    const unsigned lds0 =
        __builtin_amdgcn_readfirstlane((unsigned)(uintptr_t)&Btile[wv][0][0]);
    const unsigned lds1 =
        __builtin_amdgcn_readfirstlane((unsigned)(uintptr_t)&Btile[wv][1][0]);

    // prime the pipeline: tile 0 into buffer 0
    tdm_load_tile_f32(lds0, lKs + ((size_t)b * Nn + jbase * 16) * Dn,
                      16u, (unsigned)Dn, (unsigned)Dn);

    for (int idx = 0; idx < 8; ++idx) {
        const int cur = idx & 1;
        if (idx < 7) {
            // prior LDS reads of the buffer being overwritten must be retired
            asm volatile("s_wait_dscnt 0x0" ::: "memory");
            tdm_load_tile_f32(cur ? lds0 : lds1,
                              lKs + ((size_t)b * Nn + (jbase + idx + 1) * 16) * Dn,
                              16u, (unsigned)Dn, (unsigned)Dn);
            __builtin_amdgcn_s_wait_tensorcnt(1);   // oldest (tile idx) done
        } else {
            __builtin_amdgcn_s_wait_tensorcnt(0);   // last tile done
        }

        const float* Bl = &Btile[wv][cur][l16 * Dn];  // column n0+l16, 128 floats
        v8f acc = {};
        #pragma unroll 4
        for (int kk = 0; kk < Dn; kk += 4) {
            v2f a, bb;
            a[0] = A[l16][kk + 2 * half];
            a[1] = A[l16][kk + 2 * half + 1];
            bb   = *(const v2f*)(Bl + kk + 2 * half);
            acc = __builtin_amdgcn_wmma_f32_16x16x4_f32(
                false, a, false, bb, (short)0, acc, false, false);
        }

        if (half == 0) {                      // rows M=0..7 are real vehicles
            const int n = (jbase + idx) * 16 + l16;
            #pragma unroll
            for (int i = 0; i < Vn; ++i) {
                const int t = b * Vn + i;
                const float* np = ndf + ((size_t)t * Nn + n) * FND;
                float dyn = 0.f;
                #pragma unroll
                for (int f = 0; f < FND; ++f) dyn += np[f] * uLds[i][f];
                float val = (acc[i] + dyn) * invsD;
                val = tanhf(val) * 10.0f;
                if (!maskp[(size_t)t * Nn + n]) val = NEG_BIG;  // + log(mask)
                logits[(size_t)t * Nn + n] = val;
            }
        }
    }
}

// ---------------------------------------------------------------------------
// Kernel 5: per-batch flatten [V*N=8192] -> log_softmax, first-occurrence
// argmax, and the four scalar outputs.
// out layout: [0..B)=selected_vec, [B..2B)=selected_node, [2B..3B)=lp,
//             [3B..4B)=entropy   (all as float)
// ---------------------------------------------------------------------------
__global__ void k_select(const float* __restrict__ logits,
                         float* __restrict__ out) {
    const int b = blockIdx.x;
    const float* base = logits + (size_t)b * Vn * Nn;   // contiguous 8192
    const int tid = threadIdx.x;

    __shared__ float smax[256];
    __shared__ int   sidx[256];
    __shared__ float ssum[256];

    float mx = -3.4e38f; int mi = Vn * Nn;
    for (int i = tid; i < Vn * Nn; i += 256) {
        const float v = base[i];
        if (v > mx) { mx = v; mi = i; }          // ascending i keeps first max
    }
    smax[tid] = mx; sidx[tid] = mi;
    __syncthreads();
    for (int s = 128; s > 0; s >>= 1) {
        if (tid < s) {
            const float v2 = smax[tid + s]; const int i2 = sidx[tid + s];
            if (v2 > smax[tid] || (v2 == smax[tid] && i2 < sidx[tid])) {
                smax[tid] = v2; sidx[tid] = i2;
            }
        }
        __syncthreads();
    }
    const float gmax = smax[0];
    const int   gidx = sidx[0];

    float ls = 0.f;
    for (int i = tid; i < Vn * Nn; i += 256) ls += __expf(base[i] - gmax);
    ssum[tid] = ls;
    __syncthreads();
    for (int s = 128; s > 0; s >>= 1) {
        if (tid < s) ssum[tid] += ssum[tid + s];
        __syncthreads();
    }

    if (tid == 0) {
        const float lse  = __logf(ssum[0]);
        const float lp   = (base[gidx] - gmax) - lse;
        const float prob = __expf(lp);
        out[0 * Bn + b] = (float)gidx / (float)Nn;   // selected_vec = action / N
        out[1 * Bn + b] = (float)(gidx % Nn);        // selected_node
        out[2 * Bn + b] = lp;                        // log-prob
        out[3 * Bn + b] = -(prob * lp);              // entropy term
    }
}

// ---------------------------------------------------------------------------
// Host-side launcher. d_in order follows setup_inputs():
//  0 num_vehicles(i32)  1 node_embeddings(unused)  2 fixed_context
//  3 prev_node_embeddings(unused)  4 node_dynamic_features
//  5 vehicle_dynamic_features  6 glimpse_V_static  7 glimpse_K_static
//  8 logit_K_static  9 feasibility_mask(bool/u8)  10 W_cs  11 W_ns  12 W_out
// Workspace (floats): query[T*D] wK[T*64] heads[T*D] fQ[T*D] u[T*8]
//                     logits[T*N]  => ~1.45 MB total.
// ---------------------------------------------------------------------------
extern "C" void kernel_launch(void* const* d_in, const int* in_sizes, int n_in,
                              void* d_out, int out_size, void* d_ws, size_t ws_size,
                              hipStream_t stream) {
    const float* fc   = (const float*)d_in[2];
    const float* ndf  = (const float*)d_in[4];
    const float* vdf  = (const float*)d_in[5];
    const float* gV   = (const float*)d_in[6];
    const float* gK   = (const float*)d_in[7];
    const float* lKs  = (const float*)d_in[8];
    const unsigned char* maskp = (const unsigned char*)d_in[9];
    const float* W_cs = (const float*)d_in[10];
    const float* W_ns = (const float*)d_in[11];
    const float* W_out= (const float*)d_in[12];
    float* out = (float*)d_out;

    float* ws      = (float*)d_ws;
    float* query_w = ws;                         // T*D
    float* wK_w    = query_w + Tn * Dn;          // T*64
    float* heads_w = wK_w    + Tn * 64;          // T*D
    float* fQ_w    = heads_w + Tn * Dn;          // T*D
    float* u_w     = fQ_w    + Tn * Dn;          // T*8
    float* logit_w = u_w     + Tn * FND;         // T*N

    k_setup <<<Tn, 128, 0, stream>>>(fc, vdf, W_cs, W_ns, query_w, wK_w);
    k_attn  <<<Tn, 256, 0, stream>>>(query_w, wK_w, ndf, gK, gV, maskp, W_ns, heads_w);
    k_final <<<Tn, 128, 0, stream>>>(heads_w, W_out, W_ns, fQ_w, u_w);
    k_logits<<<Bn * 4, 64, 0, stream>>>(fQ_w, u_w, lKs, ndf, maskp, logit_w);
    k_select<<<Bn, 256, 0, stream>>>(logit_w, out);
}